// GCN_25460566131065
// MI455X (gfx1250) — compile-verified
//
#include <hip/hip_runtime.h>
#include <hip/hip_bf16.h>
#include <math.h>

typedef __attribute__((ext_vector_type(2))) float v2f;
typedef __attribute__((ext_vector_type(8))) float v8f;

#define HIDDEN 128
#define NGRAPH 256
#define WT 132   // transposed-W LDS stride (floats): 132 % 64 == 4 -> b64 B-fragment
                 // reads are bank-conflict-free across both half-waves; 8B aligned

// ---------------- degree / normalization ----------------

__global__ void deg_count_kernel(const int* __restrict__ dst,
                                 float* __restrict__ deg, int E) {
  int e = blockIdx.x * blockDim.x + threadIdx.x;
  if (e < E) atomicAdd(&deg[dst[e]], 1.0f);
}

__global__ void dis_kernel(const float* __restrict__ deg,
                           float* __restrict__ dis, int N) {
  int i = blockIdx.x * blockDim.x + threadIdx.x;
  if (i < N) dis[i] = rsqrtf(deg[i] + 1.0f);  // +1 self-loop; deg always > 0
}

// ---------------- layer 0 GEMM (K = IN_FEAT, tiny) ----------------

__global__ void gemm0_kernel(const float* __restrict__ x,
                             const float* __restrict__ W0,
                             float* __restrict__ T, int N, int K) {
  long long i = (long long)blockIdx.x * blockDim.x + threadIdx.x;
  if (i >= (long long)N * HIDDEN) return;
  int node = (int)(i >> 7);
  int f = (int)i & 127;
  const float* xp = x + (size_t)node * K;
  float s = 0.0f;
  for (int k = 0; k < K; ++k) s = fmaf(xp[k], W0[(size_t)k * HIDDEN + f], s);
  T[i] = s;
}

// ---------------- fp32 WMMA GEMM: T = A[N,128] @ W[128,128] ----------------
// W staged TRANSPOSED once per block into LDS, so each B fragment is one
// contiguous ds_load_b64 straight into the even-aligned pair WMMA wants.
// One wave per 16-row strip; 8 v8f accumulators cover the full 128-wide row.

__global__ void __launch_bounds__(256) gemm_wmma_kernel(
    const float* __restrict__ A, const float* __restrict__ W,
    float* __restrict__ T, int N) {
  __shared__ float Wt[HIDDEN * WT];   // ~68 KB of the WGP's 320 KB, Wt[n][k]

  // cooperative stage with transpose: coalesced float4 global reads
  for (int idx = threadIdx.x; idx < HIDDEN * (HIDDEN / 4); idx += 256) {
    int r = idx >> 5;          // k row of W
    int c = (idx & 31) * 4;    // n column
    float4 v = *reinterpret_cast<const float4*>(W + (size_t)r * HIDDEN + c);
    Wt[(c + 0) * WT + r] = v.x;
    Wt[(c + 1) * WT + r] = v.y;
    Wt[(c + 2) * WT + r] = v.z;
    Wt[(c + 3) * WT + r] = v.w;
  }
  __syncthreads();

  const int lane = threadIdx.x & 31;
  const int strip = blockIdx.x * 8 + (threadIdx.x >> 5);
  const int row0 = strip * 16;
  if (row0 >= N) return;

  if (row0 + 16 <= N) {
    const int m  = lane & 15;   // row (A) / col (B,C)
    const int kh = lane >> 4;   // K-half select
    v8f acc[8] = {};

    for (int k0 = 0; k0 < HIDDEN; k0 += 4) {
      // A fragment: lane holds A[row0+m][k0 + 2*kh + {0,1}]  (8B contiguous)
      const float* ap = A + (size_t)(row0 + m) * HIDDEN + k0 + kh * 2;
      v2f a; a.x = ap[0]; a.y = ap[1];
      const int kb = k0 + kh * 2;
#pragma unroll
      for (int nt = 0; nt < 8; ++nt) {
        // B fragment: lane holds W[kb + {0,1}][nt*16 + m] = Wt[n][kb..kb+1]
        const int n = nt * 16 + m;
        v2f b = *reinterpret_cast<const v2f*>(&Wt[n * WT + kb]);
        acc[nt] = __builtin_amdgcn_wmma_f32_16x16x4_f32(
            false, a, false, b, (short)0, acc[nt], false, false);
      }
    }
    // C/D layout: VGPR r -> row r + 8*kh, col = m
#pragma unroll
    for (int nt = 0; nt < 8; ++nt)
#pragma unroll
      for (int r = 0; r < 8; ++r)
        T[(size_t)(row0 + r + 8 * kh) * HIDDEN + nt * 16 + m] = acc[nt][r];
  } else {
    // scalar tail (whole-wave uniform branch; unused when N % 16 == 0)
    for (int r = row0; r < N; ++r)
      for (int f = lane; f < HIDDEN; f += 32) {
        float s = 0.0f;
        for (int k = 0; k < HIDDEN; ++k)
          s = fmaf(A[(size_t)r * HIDDEN + k], Wt[f * WT + k], s);
        T[(size_t)r * HIDDEN + f] = s;
      }
  }
}

// ---------------- edge scatter: agg[dst] += dis[s]*dis[d] * T[src] ----------------
// 32 lanes per edge, float4 gather + 4 fp32 atomics each.

__global__ void scatter_kernel(const int* __restrict__ src,
                               const int* __restrict__ dst,
                               const float* __restrict__ dis,
                               const float* __restrict__ T,
                               float* __restrict__ agg, int E) {
  long long tid = (long long)blockIdx.x * blockDim.x + threadIdx.x;
  if (tid >= (long long)E * 32) return;
  int e  = (int)(tid >> 5);
  int fc = ((int)tid & 31) * 4;
  int s = src[e], d = dst[e];
  float nm = dis[s] * dis[d];
  const float4 v = *reinterpret_cast<const float4*>(T + (size_t)s * HIDDEN + fc);
  float* op = agg + (size_t)d * HIDDEN + fc;
  atomicAdd(op + 0, nm * v.x);
  atomicAdd(op + 1, nm * v.y);
  atomicAdd(op + 2, nm * v.z);
  atomicAdd(op + 3, nm * v.w);
}

// ---------------- finalize: agg = [relu](agg + dis^2 * T + b) ----------------

__global__ void finalize_kernel(float* __restrict__ agg,
                                const float* __restrict__ T,
                                const float* __restrict__ dis,
                                const float* __restrict__ bias,
                                int N, int do_relu) {
  long long i = (long long)blockIdx.x * blockDim.x + threadIdx.x;
  if (i >= (long long)N * HIDDEN) return;
  int node = (int)(i >> 7);
  int f = (int)i & 127;
  float dv = dis[node];
  float v = agg[i] + dv * dv * T[i] + bias[f];
  if (do_relu) v = fmaxf(v, 0.0f);
  agg[i] = v;
}

// ---------------- global mean pool + head ----------------

__global__ void pool_kernel(const float* __restrict__ h,
                            const int* __restrict__ batch,
                            float* __restrict__ sums,
                            float* __restrict__ cnt, int N) {
  long long i = (long long)blockIdx.x * blockDim.x + threadIdx.x;
  if (i >= (long long)N * HIDDEN) return;
  int node = (int)(i >> 7);
  int f = (int)i & 127;
  int g = batch[node];
  atomicAdd(&sums[(size_t)g * HIDDEN + f], h[i]);
  if (f == 0) atomicAdd(&cnt[g], 1.0f);
}

__global__ void head_kernel(const float* __restrict__ sums,
                            const float* __restrict__ cnt,
                            const float* __restrict__ lw,
                            const float* __restrict__ lb,
                            float* __restrict__ out, int G) {
  int g = blockIdx.x * blockDim.x + threadIdx.x;
  if (g >= G) return;
  float c = fmaxf(cnt[g], 1.0f);
  float acc = lb[0];
  for (int f = 0; f < HIDDEN; ++f)
    acc += (sums[(size_t)g * HIDDEN + f] / c) * lw[f];
  out[g] = 1.0f / (1.0f + expf(-acc));
}

// ---------------- launch ----------------

extern "C" void kernel_launch(void* const* d_in, const int* in_sizes, int n_in,
                              void* d_out, int out_size, void* d_ws, size_t ws_size,
                              hipStream_t stream) {
  const float* x     = (const float*)d_in[0];
  const int*   eidx  = (const int*)  d_in[1];   // [2, E] (int32 under default JAX)
  const int*   batch = (const int*)  d_in[2];
  const float* W0 = (const float*)d_in[3]; const float* b0 = (const float*)d_in[4];
  const float* W1 = (const float*)d_in[5]; const float* b1 = (const float*)d_in[6];
  const float* W2 = (const float*)d_in[7]; const float* b2 = (const float*)d_in[8];
  const float* lw = (const float*)d_in[9]; const float* lb = (const float*)d_in[10];
  float* out = (float*)d_out;

  const int N   = in_sizes[2];
  const int E   = in_sizes[1] / 2;
  const int INF = in_sizes[0] / N;

  float* ws   = (float*)d_ws;
  float* deg  = ws;
  float* dis  = ws + N;
  float* bufA = ws + 2 * (size_t)N;
  float* bufB = bufA + (size_t)N * HIDDEN;
  float* sums = bufB + (size_t)N * HIDDEN;     // cnt follows sums (contiguous)

  const int* srcIdx = eidx;
  const int* dstIdx = eidx + E;

  const int TB = 256;
  const long long nh = (long long)N * HIDDEN;
  const int nhBlocks   = (int)((nh + TB - 1) / TB);
  const int scBlocks   = (int)(((long long)E * 32 + TB - 1) / TB);
  const int strips     = (N + 15) / 16;
  const int gemmBlocks = (strips + 7) / 8;

  // normalization
  hipMemsetAsync(deg, 0, (size_t)N * sizeof(float), stream);
  deg_count_kernel<<<(E + TB - 1) / TB, TB, 0, stream>>>(dstIdx, deg, E);
  dis_kernel<<<(N + TB - 1) / TB, TB, 0, stream>>>(deg, dis, N);

  // layer 0 (K = IN_FEAT, VALU GEMM)
  gemm0_kernel<<<nhBlocks, TB, 0, stream>>>(x, W0, bufB, N, INF);
  hipMemsetAsync(bufA, 0, (size_t)nh * sizeof(float), stream);
  scatter_kernel<<<scBlocks, TB, 0, stream>>>(srcIdx, dstIdx, dis, bufB, bufA, E);
  finalize_kernel<<<nhBlocks, TB, 0, stream>>>(bufA, bufB, dis, b0, N, 1);

  // layer 1 (fp32 WMMA GEMM, transposed W in LDS)
  gemm_wmma_kernel<<<gemmBlocks, TB, 0, stream>>>(bufA, W1, bufB, N);
  hipMemsetAsync(bufA, 0, (size_t)nh * sizeof(float), stream);
  scatter_kernel<<<scBlocks, TB, 0, stream>>>(srcIdx, dstIdx, dis, bufB, bufA, E);
  finalize_kernel<<<nhBlocks, TB, 0, stream>>>(bufA, bufB, dis, b1, N, 1);

  // layer 2 (fp32 WMMA GEMM, no ReLU)
  gemm_wmma_kernel<<<gemmBlocks, TB, 0, stream>>>(bufA, W2, bufB, N);
  hipMemsetAsync(bufA, 0, (size_t)nh * sizeof(float), stream);
  scatter_kernel<<<scBlocks, TB, 0, stream>>>(srcIdx, dstIdx, dis, bufB, bufA, E);
  finalize_kernel<<<nhBlocks, TB, 0, stream>>>(bufA, bufB, dis, b2, N, 0);

  // pool + head
  hipMemsetAsync(sums, 0, (size_t)(NGRAPH * HIDDEN + NGRAPH) * sizeof(float), stream);
  pool_kernel<<<nhBlocks, TB, 0, stream>>>(bufA, batch, sums, sums + NGRAPH * HIDDEN, N);
  head_kernel<<<1, NGRAPH, 0, stream>>>(sums, sums + NGRAPH * HIDDEN, lw, lb, out, NGRAPH);
}